// MLSTMlayer_7413113553437
// MI455X (gfx1250) — compile-verified
//
#include <hip/hip_runtime.h>

typedef float v2f __attribute__((ext_vector_type(2)));
typedef float v8f __attribute__((ext_vector_type(8)));

#define NDIM  32
#define TLEN  256
#define BATCH 128

__device__ __forceinline__ float sigmoidf_fast(float x) {
    return 1.0f / (1.0f + __expf(-x));
}
__device__ __forceinline__ float tanhf_fast(float x) {
    // branchless, saturates correctly at +-1 (exp overflow -> inf -> 1 - 0)
    float e = __expf(2.0f * x);
    return 1.0f - 2.0f / (e + 1.0f);
}

// D = A(16x4 f32) * B(4x16 f32) + C ; fp32 WMMA (matches reference dtype)
__device__ __forceinline__ v8f wmma_f32(v2f a, v2f b, v8f c) {
    return __builtin_amdgcn_wmma_f32_16x16x4_f32(
        /*neg_a=*/false, a, /*neg_b=*/false, b,
        /*c_mod=*/(short)0, c, /*reuse_a=*/false, /*reuse_b=*/false);
}

// One 16x16 output tile of a 32x32x32 matmul.
// A row-major in LDS (stride 32); B stored TRANSPOSED in LDS (Bt[n][k]).
// A-operand: lane%16 = M row, lane/16 selects K pair -> contiguous b64 load.
// B-operand: lane%16 = N col, lane/16 selects K pair -> contiguous b64 load.
__device__ __forceinline__ v8f mm_tile(const float* A, const float* Bt,
                                       int mi, int ni, int lm, int half, v8f acc) {
    const float* arow = A  + (mi * 16 + lm) * NDIM;
    const float* brow = Bt + (ni * 16 + lm) * NDIM;
#pragma unroll
    for (int k = 0; k < NDIM; k += 4) {
        const int ka = k + 2 * half;
        v2f a;  a.x  = arow[ka]; a.y  = arow[ka + 1];
        v2f bb; bb.x = brow[ka]; bb.y = brow[ka + 1];
        acc = wmma_f32(a, bb, acc);
    }
    return acc;
}

// C/D layout: VGPR v holds M = v (lanes 0-15) / v+8 (lanes 16-31), N = lane%16.
// Row-major store (for A-use matrices): stride-32 scalar stores.
__device__ __forceinline__ void store_tile(float* D, int mi, int ni,
                                           int lm, int half, v8f acc) {
    const int n = ni * 16 + lm;
#pragma unroll
    for (int v = 0; v < 8; ++v) {
        const int m = mi * 16 + v + 8 * half;
        D[m * NDIM + n] = acc[v];
    }
}

// Transposed store (for B-use matrices): 8 contiguous floats per lane
// (vectorizable into ds_store_b128 pairs).
__device__ __forceinline__ void store_tile_T(float* Dt, int mi, int ni,
                                             int lm, int half, v8f acc) {
    const int n = ni * 16 + lm;
    float* drow = Dt + n * NDIM + mi * 16 + 8 * half;
#pragma unroll
    for (int v = 0; v < 8; ++v) drow[v] = acc[v];
}

__global__ __launch_bounds__(256, 1)
void mlstm_wmma_kernel(const float* __restrict__ X,
                       const float* __restrict__ Hp,
                       const float* __restrict__ Cp,
                       const float* __restrict__ U1,
                       const float* __restrict__ V1,
                       const float* __restrict__ U2,
                       const float* __restrict__ V2,
                       const float* __restrict__ Bp,
                       float* __restrict__ out) {
    __shared__ float sU2[4][NDIM][NDIM];    // 16 KB  A-use, row-major
    __shared__ float sV2T[4][NDIM][NDIM];   // 16 KB  B-use, transposed [g][n][k]
    __shared__ float sU1[4][NDIM];          // 0.5 KB
    __shared__ float sV1[4][2][NDIM];       // 1 KB
    __shared__ float sHT[NDIM][NDIM];       // 4 KB   B-use, transposed [n][k]
    __shared__ float sCT[2][NDIM][NDIM];    // 8 KB   B-use, transposed, ping-pong
    __shared__ float sGA[3][NDIM][NDIM];    // 12 KB  A-use gates: 0=I, 1=F, 2=O
    __shared__ float sChatT[NDIM][NDIM];    // 4 KB   B-use, transposed Chat

    const int b    = blockIdx.x;
    const int tid  = threadIdx.x;
    const int wave = tid >> 5;
    const int lane = tid & 31;
    const int half = lane >> 4;
    const int lm   = lane & 15;

    // ---- stage weights + initial state into LDS (once) ----
    {
        float* du2 = &sU2[0][0][0];
        for (int i = tid; i < 4 * NDIM * NDIM; i += 256) {
            du2[i] = U2[i];
            const int gg = i >> 10, rr = (i >> 5) & 31, cc = i & 31;
            sV2T[gg][cc][rr] = V2[i];                      // transpose V2
        }
        const float* hp = Hp + (size_t)b * NDIM * NDIM;
        const float* cp = Cp + (size_t)b * NDIM * NDIM;
        for (int i = tid; i < NDIM * NDIM; i += 256) {
            const int rr = i >> 5, cc = i & 31;
            sHT[cc][rr]    = hp[i];                        // transpose H
            sCT[0][cc][rr] = cp[i];                        // transpose C
        }
        if (tid < 4 * NDIM) (&sU1[0][0])[tid] = U1[tid];
        if (tid < 4 * 2 * NDIM) (&sV1[0][0][0])[tid] = V1[tid];
    }
    __syncthreads();

    const int g     = wave >> 1;              // phase-A gate (2 waves per gate)
    const int mi    = wave & 1;               // phase-A M row-block
    const int gslot = (g == 3) ? 2 : g;       // row-major slot for I/F/O
    const float* BpG = Bp + g * NDIM * NDIM;
    int p = 0;

    for (int t = 0; t < TLEN; ++t) {
        const float x0 = X[((size_t)b * TLEN + t) * 2 + 0];
        const float x1 = X[((size_t)b * TLEN + t) * 2 + 1];

        // ---------- Phase A: G[g] = act(U2@H@V2 + U1*(x*V1) + Bp) ----------
        {
            // Row-major T1 scratch: own sGA slot for I/F/O; gate 2 borrows the
            // dead phase-B target sCT[1-p] (disjoint rows per wave, overwritten
            // entirely by phase B after the barrier).
            float* scratch = (g == 2) ? &sCT[1 - p][0][0] : &sGA[gslot][0][0];

            // T1 = U2[g] @ H  (rows mi*16..mi*16+15)
            for (int ni = 0; ni < 2; ++ni) {
                v8f acc = {};
                acc = mm_tile(&sU2[g][0][0], &sHT[0][0], mi, ni, lm, half, acc);
                store_tile(scratch, mi, ni, lm, half, acc);
            }
            // G = T1 @ V2[g] + U1[g] w^T + Bp[g]; activation
            v8f r[2];
            for (int ni = 0; ni < 2; ++ni) {
                const int n = ni * 16 + lm;
                const float w = x0 * sV1[g][0][n] + x1 * sV1[g][1][n];
                v8f acc;
#pragma unroll
                for (int v = 0; v < 8; ++v) {
                    const int m = mi * 16 + v + 8 * half;
                    acc[v] = BpG[m * NDIM + n] + sU1[g][m] * w;
                }
                acc = mm_tile(scratch, &sV2T[g][0][0], mi, ni, lm, half, acc);
#pragma unroll
                for (int v = 0; v < 8; ++v)
                    acc[v] = (g == 2) ? tanhf_fast(acc[v]) : sigmoidf_fast(acc[v]);
                r[ni] = acc;
            }
            for (int ni = 0; ni < 2; ++ni) {
                if (g == 2) store_tile_T(&sChatT[0][0], mi, ni, lm, half, r[ni]);
                else        store_tile(&sGA[gslot][0][0], mi, ni, lm, half, r[ni]);
            }
        }
        __syncthreads();

        // ---------- Phase B: C_new = F@C + I@Chat (waves 0-3, one tile each) ----------
        if (wave < 4) {
            const int bmi = wave >> 1, bni = wave & 1;
            v8f acc = {};
            acc = mm_tile(&sGA[1][0][0], &sCT[p][0][0], bmi, bni, lm, half, acc); // F@C
            acc = mm_tile(&sGA[0][0][0], &sChatT[0][0], bmi, bni, lm, half, acc); // I@Chat
            store_tile_T(&sCT[1 - p][0][0], bmi, bni, lm, half, acc);
        }
        __syncthreads();

        // ---------- Phase C: H_new = O @ tanh(C_new) ----------
        if (wave < 4) {
            const int cmi = wave >> 1, cni = wave & 1;
            const int n = cni * 16 + lm;
            const float* arow = &sGA[2][0][0] + (cmi * 16 + lm) * NDIM;  // O
            const float* brow = &sCT[1 - p][0][0] + n * NDIM;            // C_new^T
            v8f acc = {};
#pragma unroll
            for (int k = 0; k < NDIM; k += 4) {
                const int ka = k + 2 * half;
                v2f a;  a.x  = arow[ka];             a.y  = arow[ka + 1];
                v2f bb; bb.x = tanhf_fast(brow[ka]); bb.y = tanhf_fast(brow[ka + 1]);
                acc = wmma_f32(a, bb, acc);
            }
            store_tile_T(&sHT[0][0], cmi, cni, lm, half, acc);
            float* oH = out + ((size_t)b * TLEN + t) * (NDIM * NDIM);
#pragma unroll
            for (int v = 0; v < 8; ++v) {
                const int m = cmi * 16 + v + 8 * half;
                oH[m * NDIM + n] = acc[v];
            }
        }
        __syncthreads();
        p ^= 1;
    }

    // ---- final states: H_T then C_T (un-transpose on the way out) ----
    {
        float* oH = out + (size_t)BATCH * TLEN * (NDIM * NDIM) + (size_t)b * (NDIM * NDIM);
        float* oC = oH + (size_t)BATCH * (NDIM * NDIM);
        const float* h = &sHT[0][0];
        const float* c = &sCT[p][0][0];
        for (int i = tid; i < NDIM * NDIM; i += 256) {
            const int rr = i >> 5, cc = i & 31;
            oH[i] = h[cc * NDIM + rr];
            oC[i] = c[cc * NDIM + rr];
        }
    }
}

extern "C" void kernel_launch(void* const* d_in, const int* in_sizes, int n_in,
                              void* d_out, int out_size, void* d_ws, size_t ws_size,
                              hipStream_t stream) {
    const float* X  = (const float*)d_in[0];
    const float* Hp = (const float*)d_in[1];
    const float* Cp = (const float*)d_in[2];
    const float* U1 = (const float*)d_in[3];
    const float* V1 = (const float*)d_in[4];
    const float* U2 = (const float*)d_in[5];
    const float* V2 = (const float*)d_in[6];
    const float* Bp = (const float*)d_in[7];
    float* out = (float*)d_out;
    mlstm_wmma_kernel<<<BATCH, 256, 0, stream>>>(X, Hp, Cp, U1, V1, U2, V2, Bp, out);
}